// QuantileNormalizer_61572651155629
// MI455X (gfx1250) — compile-verified
//
#include <hip/hip_runtime.h>
#include <stdint.h>

#define N_Q     1000
#define N_FEAT  32
#define BATCH   4096
#define EPS_F   1e-9f

typedef __attribute__((ext_vector_type(4))) unsigned int u32x4;
typedef __attribute__((ext_vector_type(4))) int          i32x4;
typedef __attribute__((ext_vector_type(8))) int          i32x8;

// Issue one TDM load of a 2-D tile (tile_d0 x tile_d1 elements of 4 bytes)
// from global memory into LDS at byte offset lds_off.
// Descriptor layout per CDNA5 ISA section 8 (Tensor DMA Descriptor).
__device__ __forceinline__ void tdm_load_2d_f32(uint32_t lds_off, const void* gptr,
                                                uint32_t tensor_d0, uint32_t tensor_d1,
                                                uint32_t tile_d0,   uint32_t tile_d1,
                                                uint32_t d0_stride)
{
    uint64_t ga = (uint64_t)(uintptr_t)gptr;

    u32x4 g0;
    g0[0] = 1u;                                             // count=1 (valid), user mode, no gather
    g0[1] = lds_off;                                        // lds_addr (bytes)
    g0[2] = (uint32_t)(ga & 0xFFFFFFFFu);                   // global_addr[31:0]
    g0[3] = (uint32_t)((ga >> 32) & 0x1FFFFFFu)             // global_addr[56:32]
          | (2u << 30);                                     // type = 2 ("image")

    i32x8 g1;
    g1[0] = (int)(2u << 16);                                // data_size=2 (4 bytes); no multicast/pad/iterate
    g1[1] = (int)((tensor_d0 & 0xFFFFu) << 16);             // tensor_dim0[15:0] @ bits 63:48
    g1[2] = (int)(((tensor_d0 >> 16) & 0xFFFFu)             // tensor_dim0[31:16]
          |       ((tensor_d1 & 0xFFFFu) << 16));           // tensor_dim1[15:0]
    g1[3] = (int)(((tensor_d1 >> 16) & 0xFFFFu)             // tensor_dim1[31:16]
          |       ((tile_d0 & 0xFFFFu) << 16));             // tile_dim0
    g1[4] = (int)(tile_d1 & 0xFFFFu);                       // tile_dim1 (tile_dim2 = 0 -> unused)
    g1[5] = (int)d0_stride;                                 // tensor_dim0_stride[31:0]
    g1[6] = 0;                                              // stride[47:32]=0, dim1_stride lo = 0
    g1[7] = 0;                                              // dim1_stride hi = 0

    i32x4 gz = {0, 0, 0, 0};                                // groups 2/3: 2-D tensor, unused dims
#if defined(__clang_major__) && (__clang_major__ >= 23)
    i32x8 gz8 = {0, 0, 0, 0, 0, 0, 0, 0};
    __builtin_amdgcn_tensor_load_to_lds(g0, g1, gz, gz, gz8, 0);
#else
    __builtin_amdgcn_tensor_load_to_lds(g0, g1, gz, gz, 0);
#endif
}

__global__ __launch_bounds__(256)
void qnorm_kernel(const float* __restrict__ x,
                  const float* __restrict__ q_values,   // (N_Q, N_FEAT) row-major
                  const float* __restrict__ quantiles,  // (N_Q,)
                  float* __restrict__ out)              // (BATCH, N_FEAT)
{
    __shared__ float s_q[N_Q];   // q_values[:, f]  (sorted ascending)
    __shared__ float s_t[N_Q];   // quantiles[:]

    const int f = blockIdx.x;                              // feature
    const int b = blockIdx.y * blockDim.x + threadIdx.x;   // batch row

    if (threadIdx.x == 0) {
        // Generic->LDS offset: low 32 bits of the shared aperture address are
        // the allocation-relative LDS byte address (ISA 10.2).
        uint32_t off_q = (uint32_t)(uintptr_t)(void*)s_q;
        uint32_t off_t = (uint32_t)(uintptr_t)(void*)s_t;
        // Column f of q_values: 1 x N_Q tile, row stride N_FEAT elements.
        tdm_load_2d_f32(off_q, q_values + f, N_FEAT, N_Q, 1, N_Q, N_FEAT);
        // quantiles: contiguous N_Q x 1 tile.
        tdm_load_2d_f32(off_t, quantiles, N_Q, 1, N_Q, 1, N_Q);
    }
    __builtin_amdgcn_s_wait_tensorcnt(0);   // DMA-issuing wave waits; others no-op
    __syncthreads();                        // orders LDS reads after DMA completion

    const float xv = x[b * N_FEAT + f];

    // upper_bound: first idx in [0, N_Q] with s_q[idx] > xv.
    // 1000 <= 2^10, so 10 fixed branchless bisection steps converge exactly.
    int lo = 0, hi = N_Q;
#pragma unroll
    for (int it = 0; it < 10; ++it) {
        int mid = (lo + hi) >> 1;
        bool le = (s_q[mid] <= xv);
        lo = le ? (mid + 1) : lo;
        hi = le ? hi        : mid;
    }
    int rank = lo - 1;                                  // == sum(x >= q) - 1
    rank = rank < 0 ? 0 : (rank > (N_Q - 2) ? (N_Q - 2) : rank);

    const float low  = s_q[rank];
    const float high = s_q[rank + 1];
    const float ql   = s_t[rank];
    const float qh   = s_t[rank + 1];

    const float t = (xv - low) / (high - low + EPS_F);
    out[b * N_FEAT + f] = ql + t * (qh - ql);
}

extern "C" void kernel_launch(void* const* d_in, const int* in_sizes, int n_in,
                              void* d_out, int out_size, void* d_ws, size_t ws_size,
                              hipStream_t stream) {
    const float* x         = (const float*)d_in[0];   // (BATCH, N_FEAT)
    const float* q_values  = (const float*)d_in[1];   // (N_Q, N_FEAT)
    const float* quantiles = (const float*)d_in[2];   // (N_Q,)
    float* out             = (float*)d_out;

    dim3 grid(N_FEAT, BATCH / 256);
    qnorm_kernel<<<grid, dim3(256), 0, stream>>>(x, q_values, quantiles, out);
}